// CombinedRankingLoss_52355651338505
// MI455X (gfx1250) — compile-verified
//
#include <hip/hip_runtime.h>
#include <hip/hip_bf16.h>
#include <cmath>

// CombinedRankingLoss for MI455X (gfx1250, wave32).
// Pairwise rank loss tile via V_WMMA_F32_16X16X4_F32:
//   masked_diff[i][j] = pos_i*neg_j*(MARGIN - lgt_i + lgt_j)
//                     = [pos_i*(MARGIN-lgt_i)]*[neg_j] + [pos_i]*[neg_j*lgt_j]
// rank-2 -> one WMMA per 16x16 tile; relu folded as (x+|x|)/2 into the FMA
// accumulator (|x| is a free VALU source modifier).

#define MARGIN        0.5f
#define LAMBDA_RANK   1.0f
#define B_BATCH       64
#define N_CAND        1024
#define N_TILES       (N_CAND / 16)    // 64 tile rows/cols
#define SPLIT         4                // blocks per batch (occupancy)
#define ROWS_PER_BLK  (N_TILES / SPLIT) // 16 tile rows per block
#define BLOCK_THREADS 256              // 8 waves of 32
#define WAVES         (BLOCK_THREADS / 32)

typedef __attribute__((ext_vector_type(2))) float v2f;
typedef __attribute__((ext_vector_type(8))) float v8f;

__global__ __launch_bounds__(BLOCK_THREADS)
void rank_batch_kernel(const float* __restrict__ logits,
                       const float* __restrict__ labels,
                       const float* __restrict__ weights,
                       float* __restrict__ ws)
{
    // Packed rank-2 operands: one ds_load_b64 per fragment half.
    __shared__ v2f s_a2[N_CAND];   // { pos_i*(MARGIN-lgt_i), pos_i }      (A: K=0,K=1)
    __shared__ v2f s_b2[N_CAND];   // { neg_j,               neg_j*lgt_j } (B: K=0,K=1)
    __shared__ float s_bce, s_npos, s_nneg, s_total;

    const int blk  = blockIdx.x;
    const int b    = blk / SPLIT;        // batch row
    const int part = blk % SPLIT;        // which quarter of tile rows / elements
    const int tid  = threadIdx.x;

    if (tid == 0) { s_bce = 0.f; s_npos = 0.f; s_nneg = 0.f; s_total = 0.f; }
    __syncthreads();

    // ---- Phase 1: build operands (full row), stats over this block's quarter ----
    const int base = b * N_CAND;
    float bce_acc = 0.f, pos_acc = 0.f, neg_acc = 0.f;
#pragma unroll
    for (int k = 0; k < N_CAND / BLOCK_THREADS; ++k) {
        const int n = tid + k * BLOCK_THREADS;
        float x = logits [base + n];
        float y = labels [base + n];
        float p = (y == 1.f) ? 1.f : 0.f;
        float g = (y == 0.f) ? 1.f : 0.f;
        v2f av; av.x = p * (MARGIN - x); av.y = p;
        v2f bv; bv.x = g;                bv.y = g * x;
        s_a2[n] = av;
        s_b2[n] = bv;
        if (k == part) {   // uniform branch: stats only for this block's quarter
            float w = weights[base + n];
            bce_acc += w * (fmaxf(x, 0.f) - x * y + log1pf(__expf(-fabsf(x))));
            pos_acc += p;
            neg_acc += g;
        }
    }
    atomicAdd(&s_bce,  bce_acc);    // ds_add_f32, block-local
    atomicAdd(&s_npos, pos_acc);
    atomicAdd(&s_nneg, neg_acc);
    __syncthreads();

    // ---- Phase 2: this block's 16x64 grid of 16x16 WMMA tiles ----
    const int wave = tid >> 5;          // wave32
    const int lane = tid & 31;
    const int m    = lane & 15;
    // Lanes 16-31 carry K=2,3 of the fragments -> zero via mask multiply
    // (no EXEC predication; keeps EXEC all-1s and avoids s_and_saveexec chains).
    const float msk = (lane < 16) ? 1.f : 0.f;

    v8f acc = {};                       // relu'd pair-loss accumulator
    for (int ti = part * ROWS_PER_BLK + wave;
         ti < (part + 1) * ROWS_PER_BLK; ti += WAVES) {
        const v2f arow = s_a2[ti * 16 + m];
        v2f a; a.x = arow.x * msk; a.y = arow.y * msk;
#pragma unroll 4
        for (int tj = 0; tj < N_TILES; ++tj) {
            const v2f brow = s_b2[tj * 16 + m];   // ds_load_b64, all lanes
            v2f bm; bm.x = brow.x * msk; bm.y = brow.y * msk;
            v8f cz = {};
            // (neg_a, A, neg_b, B, c_mod, C, reuse_a, reuse_b)
            v8f d = __builtin_amdgcn_wmma_f32_16x16x4_f32(
                        false, a, false, bm, (short)0, cz, false, false);
#pragma unroll
            for (int k = 0; k < 8; ++k)   // relu(x) = 0.5x + 0.5|x|, fused accumulate
                acc[k] = fmaf(0.5f, fabsf(d[k]), fmaf(0.5f, d[k], acc[k]));
        }
    }

    float lane_sum = 0.f;
#pragma unroll
    for (int k = 0; k < 8; ++k) lane_sum += acc[k];
    atomicAdd(&s_total, lane_sum);
    __syncthreads();

    // ---- Phase 3: per-block partials (pure writes -> deterministic) ----
    if (tid == 0) {
        float* o = ws + 4 * blk;
        o[0] = s_total;   // rank-loss partial (this block's tile rows)
        o[1] = s_bce;     // BCE partial (this block's element quarter)
        o[2] = s_npos;    // pos count partial
        o[3] = s_nneg;    // neg count partial
    }
}

__global__ void finalize_kernel(const float* __restrict__ ws, float* __restrict__ out)
{
    __shared__ float s_mean[B_BATCH], s_valid[B_BATCH], s_bce[B_BATCH];
    const int t = threadIdx.x;
    if (t < B_BATCH) {
        float rank_sum = 0.f, bce = 0.f, npos = 0.f, nneg = 0.f;
#pragma unroll
        for (int p = 0; p < SPLIT; ++p) {
            const float* o = ws + 4 * (t * SPLIT + p);
            rank_sum += o[0]; bce += o[1]; npos += o[2]; nneg += o[3];
        }
        float n_pairs = npos * nneg;
        bool  valid   = n_pairs > 0.f;
        s_mean [t] = valid ? (rank_sum / fmaxf(n_pairs, 1.f)) : 0.f;
        s_valid[t] = valid ? 1.f : 0.f;
        s_bce  [t] = bce;
    }
    __syncthreads();
    if (t == 0) {
        float mean_sum = 0.f, valid_cnt = 0.f, bce_sum = 0.f;
        for (int b = 0; b < B_BATCH; ++b) {
            mean_sum += s_mean[b]; valid_cnt += s_valid[b]; bce_sum += s_bce[b];
        }
        float rank = (valid_cnt > 0.f) ? (mean_sum / fmaxf(valid_cnt, 1.f)) : 0.f;
        out[0] = bce_sum / (float)(B_BATCH * N_CAND) + LAMBDA_RANK * rank;
    }
}

extern "C" void kernel_launch(void* const* d_in, const int* in_sizes, int n_in,
                              void* d_out, int out_size, void* d_ws, size_t ws_size,
                              hipStream_t stream)
{
    const float* logits  = (const float*)d_in[0];
    const float* labels  = (const float*)d_in[1];
    const float* weights = (const float*)d_in[2];
    float*       out     = (float*)d_out;
    float*       ws      = (float*)d_ws;   // 4 * 256 floats used

    rank_batch_kernel<<<B_BATCH * SPLIT, BLOCK_THREADS, 0, stream>>>(
        logits, labels, weights, ws);
    finalize_kernel<<<1, B_BATCH, 0, stream>>>(ws, out);
}